// DotProductPredictor_14946486190729
// MI455X (gfx1250) — compile-verified
//
#include <hip/hip_runtime.h>

// u_mul_v edge kernel for MI455X (gfx1250, wave32).
// score[e,:] = h[src[e],:] * h[dst[e],:],  E=640000, D=128, fp32.
//
// Store-bandwidth bound (~328 MB out). Strategy:
//  - one wave32 per edge; lane owns a float4 (32 lanes x 16B = one 512B row)
//  - double-buffered async row gathers into LDS (ASYNCcnt pipeline)
//  - non-temporal b128 stores so the streamed output does not evict the
//    5 MB L2-resident feature table.

typedef float v4f __attribute__((ext_vector_type(4)));
typedef int   v4i __attribute__((ext_vector_type(4)));

// Exact parameter types for __builtin_amdgcn_global_load_async_to_lds_b128:
//   (v4i addrspace(1)*, v4i addrspace(3)*, imm offset, imm cpol)
typedef __attribute__((address_space(1))) v4i gv4i;
typedef __attribute__((address_space(3))) v4i lv4i;

#define D_FEAT 128
#define LANES 32
#define WAVES_PER_BLOCK 8
#define BLOCK_THREADS (WAVES_PER_BLOCK * LANES)
#define ROW_FLOATS D_FEAT
#define STAGE_FLOATS (2 * ROW_FLOATS)           // src row + dst row = 1 KB
#define STAGES 2
#define WAVE_LDS_FLOATS (STAGES * STAGE_FLOATS) // 2 KB per wave
#define BLOCK_LDS_FLOATS (WAVES_PER_BLOCK * WAVE_LDS_FLOATS)
#define MAX_BLOCKS 4096

__device__ __forceinline__ void async_row_b128(const float* gsrc, float* lds_dst) {
  // 32 lanes x 16B each: one full 512B feature row per instruction.
  __builtin_amdgcn_global_load_async_to_lds_b128(
      (gv4i*)gsrc, (lv4i*)lds_dst, /*imm offset*/ 0, /*cpol*/ 0);
}

__global__ __launch_bounds__(BLOCK_THREADS) void edge_umulv_kernel(
    const float* __restrict__ h,
    const int* __restrict__ src,
    const int* __restrict__ dst,
    float* __restrict__ out,
    int n_edges) {
  __shared__ float smem[BLOCK_LDS_FLOATS];

  const int lane        = threadIdx.x & (LANES - 1);
  const int wave_in_blk = threadIdx.x >> 5;
  const int wave        = blockIdx.x * WAVES_PER_BLOCK + wave_in_blk;
  const int n_waves     = gridDim.x * WAVES_PER_BLOCK;

  float* wbase = &smem[wave_in_blk * WAVE_LDS_FLOATS];

  int e = wave;
  if (e >= n_edges) return;

  // Prologue: stage 0 gathers for the first edge this wave owns.
  {
    const int is = src[e];
    const int id = dst[e];
    async_row_b128(h + (size_t)is * D_FEAT + lane * 4, wbase + lane * 4);
    async_row_b128(h + (size_t)id * D_FEAT + lane * 4, wbase + ROW_FLOATS + lane * 4);
  }

  int stage = 0;
  for (; e < n_edges; e += n_waves) {
    const int en = e + n_waves;
    if (en < n_edges) {
      // Kick off next edge's gathers into the other stage, then wait until
      // only those 2 remain in flight (async loads complete in order, so
      // ASYNCcnt<=2 means the current stage's rows have landed in LDS).
      const int is = src[en];
      const int id = dst[en];
      float* nb = wbase + (stage ^ 1) * STAGE_FLOATS;
      async_row_b128(h + (size_t)is * D_FEAT + lane * 4, nb + lane * 4);
      async_row_b128(h + (size_t)id * D_FEAT + lane * 4, nb + ROW_FLOATS + lane * 4);
      __builtin_amdgcn_s_wait_asynccnt(2);
    } else {
      __builtin_amdgcn_s_wait_asynccnt(0);
    }
    // Keep the LDS reads below the async wait at the IR level.
    __asm__ volatile("" ::: "memory");

    const float* cb = wbase + stage * STAGE_FLOATS;
    v4f a = *(const v4f*)(cb + lane * 4);               // ds_load_b128
    v4f b = *(const v4f*)(cb + ROW_FLOATS + lane * 4);  // ds_load_b128
    v4f r = a * b;

    // Streaming output: never re-read -> non-temporal, protect L2-resident h.
    __builtin_nontemporal_store(r, (v4f*)(out + (size_t)e * D_FEAT + lane * 4));

    stage ^= 1;
  }
}

extern "C" void kernel_launch(void* const* d_in, const int* in_sizes, int n_in,
                              void* d_out, int out_size, void* d_ws, size_t ws_size,
                              hipStream_t stream) {
  const float* h   = (const float*)d_in[0];
  const int*   src = (const int*)d_in[1];
  const int*   dst = (const int*)d_in[2];
  float*       out = (float*)d_out;
  const int n_edges = in_sizes[1];

  int blocks = (n_edges + WAVES_PER_BLOCK - 1) / WAVES_PER_BLOCK;
  if (blocks > MAX_BLOCKS) blocks = MAX_BLOCKS;
  if (blocks < 1) blocks = 1;

  edge_umulv_kernel<<<blocks, BLOCK_THREADS, 0, stream>>>(h, src, dst, out, n_edges);
}